// MultiHeadAttention_25202868093524
// MI455X (gfx1250) — compile-verified
//
#include <hip/hip_runtime.h>

typedef float v2f __attribute__((ext_vector_type(2)));
typedef float v8f __attribute__((ext_vector_type(8)));

// Problem dims (fixed by the reference)
constexpr int B_ = 8, C_ = 256, H_ = 8, D_ = 32;
constexpr int L_ = 1024;          // P*T
constexpr int SPLIT = 4;          // workgroups per (b,h)
constexpr int WAVES = 16;         // waves per workgroup (512 threads)

// Padded LDS strides (floats); 33, 17 coprime with 64 banks; 260 -> 4*lane banks
constexpr int ZS = 33;            // z row stride
constexpr int WS = 260;           // W row stride
constexpr int Z_FLOATS  = L_ * ZS;            // 33792
constexpr int W_FLOATS  = D_ * WS;            // 8320
constexpr int PBUF      = 16 * 17;            // per-wave P tile
constexpr int STBUF     = 32 * 17;            // per-wave output-transpose tile
constexpr int OVER_ATTN = WAVES * (PBUF + STBUF);
constexpr int OVERLAY   = (W_FLOATS > OVER_ATTN) ? W_FLOATS : OVER_ATTN;
constexpr int SMEM_FLOATS = Z_FLOATS + OVERLAY;   // 46848 floats = 183 KB < 320 KB

static __device__ __forceinline__ v8f wmma_f32(v2f a, v2f b, v8f c) {
  return __builtin_amdgcn_wmma_f32_16x16x4_f32(false, a, false, b, (short)0, c,
                                               false, false);
}

__global__ __launch_bounds__(512, 1)
void fused_mha_f32(const float* __restrict__ x, const float* __restrict__ W,
                   const float* __restrict__ bias, float* __restrict__ out) {
  __shared__ float smem[SMEM_FLOATS];
  float* zl  = smem;                       // z: [L_][ZS], z[l*ZS + d]
  float* wl  = smem + Z_FLOATS;            // projection phase: W_h [32][WS]
  float* pl  = smem + Z_FLOATS;            // attention phase: P tiles (overlays wl)
  float* stl = smem + Z_FLOATS + WAVES * PBUF; // attention phase: store staging

  const int bh   = blockIdx.x / SPLIT;
  const int spl  = blockIdx.x % SPLIT;
  const int b    = bh / H_;
  const int h    = bh % H_;
  const int wave = threadIdx.x >> 5;
  const int lane = threadIdx.x & 31;
  const int ln   = lane & 15;              // column / M-within-half index
  const int half = lane >> 4;              // lane half selects K pair / M+8

  // ---------------- stage W_h into padded LDS ----------------
  {
    const float* wg = W + (size_t)h * (D_ * C_);
    for (int i = threadIdx.x; i < D_ * C_; i += 512) {
      int d = i >> 8, c = i & 255;
      wl[d * WS + c] = wg[i];
    }
  }
  __syncthreads();

  // ---------------- projection: z[l,d] = sum_c x[b,c,l] * W[h,d,c] + bias ----
  {
    const float* xb = x + (size_t)b * (C_ * L_);
    const float bias0 = bias[h * D_ + ln];
    const float bias1 = bias[h * D_ + 16 + ln];
    for (int t = 0; t < 4; ++t) {
      const int l0 = (wave * 4 + t) * 16;
      v8f acc0 = {};
      v8f acc1 = {};
      for (int c0 = 0; c0 < C_; c0 += 4) {
        const int c = c0 + 2 * half;
        v2f a;                                  // A = x^T tile (M=l, K=c)
        a.x = xb[c * L_ + l0 + ln];
        a.y = xb[(c + 1) * L_ + l0 + ln];
        v2f b0, b1;                             // B = W^T tile (K=c, N=d)
        b0.x = wl[ln * WS + c];
        b0.y = wl[ln * WS + c + 1];
        b1.x = wl[(16 + ln) * WS + c];
        b1.y = wl[(16 + ln) * WS + c + 1];
        acc0 = wmma_f32(a, b0, acc0);
        acc1 = wmma_f32(a, b1, acc1);
      }
#pragma unroll
      for (int r = 0; r < 8; ++r) {
        const int row = l0 + r + 8 * half;      // C/D: M = r + 8*half
        zl[row * ZS + ln]      = acc0[r] + bias0;
        zl[row * ZS + 16 + ln] = acc1[r] + bias1;
      }
    }
  }
  __syncthreads();

  // ---------------- flash attention: one 16-row query tile per wave --------
  {
    const int l0 = (spl * WAVES + wave) * 16;   // query tile base
    v2f aq[8];                                  // Q in A layout, K over d=32
#pragma unroll
    for (int f = 0; f < 8; ++f) {
      const int d = 4 * f + 2 * half;
      aq[f].x = zl[(l0 + ln) * ZS + d];
      aq[f].y = zl[(l0 + ln) * ZS + d + 1];
    }
    v8f o0 = {}, o1 = {};                       // O accum (d 0..15 / 16..31)
    float rm[8], rl[8];
#pragma unroll
    for (int r = 0; r < 8; ++r) { rm[r] = -1e30f; rl[r] = 0.f; }
    const float scale = 0.17677669529663687f;   // 1/sqrt(32)
    float* pw = pl + wave * PBUF;

    for (int m0 = 0; m0 < L_; m0 += 16) {
      // S = Q * K^T  (B fragment: K index = d, N index = key m)
      v8f s = {};
#pragma unroll
      for (int f = 0; f < 8; ++f) {
        const int d = 4 * f + 2 * half;
        v2f bk;
        bk.x = zl[(m0 + ln) * ZS + d];
        bk.y = zl[(m0 + ln) * ZS + d + 1];
        s = wmma_f32(aq[f], bk, s);
      }
      // online softmax per row (row = r + 8*half; 16 lanes of its half)
      float p[8];
#pragma unroll
      for (int r = 0; r < 8; ++r) {
        const float v = s[r] * scale;
        float mx = v;
        mx = fmaxf(mx, __shfl_xor(mx, 1));
        mx = fmaxf(mx, __shfl_xor(mx, 2));
        mx = fmaxf(mx, __shfl_xor(mx, 4));
        mx = fmaxf(mx, __shfl_xor(mx, 8));
        const float nm    = fmaxf(rm[r], mx);
        const float alpha = __expf(rm[r] - nm);
        const float pv    = __expf(v - nm);
        float sum = pv;
        sum += __shfl_xor(sum, 1);
        sum += __shfl_xor(sum, 2);
        sum += __shfl_xor(sum, 4);
        sum += __shfl_xor(sum, 8);
        rl[r] = rl[r] * alpha + sum;
        rm[r] = nm;
        o0[r] *= alpha;
        o1[r] *= alpha;
        p[r] = pv;
      }
      // transpose P (C/D layout -> A layout) through per-wave LDS tile
#pragma unroll
      for (int r = 0; r < 8; ++r)
        pw[(r + 8 * half) * 17 + ln] = p[r];
      asm volatile("s_wait_dscnt 0x0" ::: "memory");
      // O += P * V   (V = same z tile; B fragment: K = key m, N = d)
#pragma unroll
      for (int g = 0; g < 4; ++g) {
        const int k = 4 * g + 2 * half;
        v2f ap;
        ap.x = pw[ln * 17 + k];
        ap.y = pw[ln * 17 + k + 1];
        v2f bv0, bv1;
        bv0.x = zl[(m0 + k) * ZS + ln];
        bv0.y = zl[(m0 + k + 1) * ZS + ln];
        bv1.x = zl[(m0 + k) * ZS + 16 + ln];
        bv1.y = zl[(m0 + k + 1) * ZS + 16 + ln];
        o0 = wmma_f32(ap, bv0, o0);
        o1 = wmma_f32(ap, bv1, o1);
      }
    }
    // normalize rows
#pragma unroll
    for (int r = 0; r < 8; ++r) {
      const float inv = 1.0f / rl[r];
      o0[r] *= inv;
      o1[r] *= inv;
    }
    // transpose to channels-first through LDS, then coalesced global stores
    float* sw = stl + wave * STBUF;             // [32 d][17]
#pragma unroll
    for (int r = 0; r < 8; ++r) {
      const int lrow = r + 8 * half;
      sw[ln * 17 + lrow]        = o0[r];
      sw[(16 + ln) * 17 + lrow] = o1[r];
    }
    asm volatile("s_wait_dscnt 0x0" ::: "memory");
    float* ob = out + (size_t)(b * H_ + h) * (D_ * L_);
#pragma unroll
    for (int rr = 0; rr < 16; ++rr) {
      const int d = rr * 2 + half;
      ob[d * L_ + l0 + ln] = sw[d * 17 + ln];
    }
  }
}

extern "C" void kernel_launch(void* const* d_in, const int* in_sizes, int n_in,
                              void* d_out, int out_size, void* d_ws, size_t ws_size,
                              hipStream_t stream) {
  (void)in_sizes; (void)n_in; (void)out_size; (void)d_ws; (void)ws_size;
  const float* x    = (const float*)d_in[0];
  const float* W    = (const float*)d_in[1];
  const float* bias = (const float*)d_in[2];
  float* out        = (float*)d_out;
  dim3 grid(B_ * H_ * SPLIT);   // 256 workgroups
  dim3 block(512);              // 16 wave32s
  hipLaunchKernelGGL(fused_mha_f32, grid, block, 0, stream, x, W, bias, out);
}